// ClustGeoEdgeEncoder_16441134809144
// MI455X (gfx1250) — compile-verified
//
#include <hip/hip_runtime.h>
#include <hip/hip_bf16.h>

typedef float v2f __attribute__((ext_vector_type(2)));
typedef float v8f __attribute__((ext_vector_type(8)));

#define KC  256   // cluster size
#define TPB 512   // 16 waves of 32

// Tile convention (transposed vs round 1):
//   tile rows M = cluster-2 voxel j, tile cols N = cluster-1 voxel i.
//   A[m,k] = (-2*x2, -2*y2, -2*z2, n2)   (row-varying norm folded into K slot)
//   B[k,n] = ( x1,    y1,    z1,   1 )
//   => (A*B)[m,n] = -2*dot(p2_j, p1_i) + n2_j ; add n1_i (lane-constant) -> d^2
// flat argmin index = i*256 + j, monotone in (ct outer, v inner) per lane.
__global__ __launch_bounds__(TPB) void clust_geo_edge_kernel(
    const float* __restrict__ data,        // [N_VOX, 5]
    const int*   __restrict__ clusts,      // [N_CLUST, 256]
    const int*   __restrict__ edge_index,  // [2, E]
    float*       __restrict__ out,         // [E, 19]
    int n_edge)
{
    __shared__ float  x1x[KC], x1y[KC], x1z[KC];
    __shared__ float  x2x[KC], x2y[KC], x2z[KC];
    __shared__ float4 bpack[16 * 32];      // B frag (.xy) + n1col (.z), per (ct, lane)
    __shared__ float2 apack[16 * 32];      // A frag, per (row-tile, lane)
    __shared__ unsigned long long wbest[16];

    const int e = blockIdx.x;
    if (e >= n_edge) return;

    const int c1 = edge_index[e];
    const int c2 = edge_index[n_edge + e];
    const int t  = threadIdx.x;

    // ---- gather stage: fetch 2x256 voxels, pre-pack WMMA fragments into LDS
    {
        const int which = t >> 8;          // 0 -> cluster 1 (B side), 1 -> cluster 2 (A side)
        const int k     = t & 255;
        const int c     = which ? c2 : c1;
        const int vox   = clusts[c * KC + k];
        const float x = data[vox * 5 + 0];
        const float y = data[vox * 5 + 1];
        const float z = data[vox * 5 + 2];
        const float n = x * x + y * y + z * z;
        const int ct = k >> 4;             // tile index
        const int m  = k & 15;             // position within tile
        if (which == 0) {
            x1x[k] = x; x1y[k] = y; x1z[k] = z;
            // B layout: lanes 0-15 hold K=0,1 ; lanes 16-31 hold K=2,3
            bpack[ct * 32 + m]      = make_float4(x, y,    n, 0.0f);
            bpack[ct * 32 + m + 16] = make_float4(z, 1.0f, n, 0.0f);
        } else {
            x2x[k] = x; x2y[k] = y; x2z[k] = z;
            // A layout: lanes 0-15 hold K=0,1 ; lanes 16-31 hold K=2,3
            apack[ct * 32 + m]      = make_float2(-2.0f * x, -2.0f * y);
            apack[ct * 32 + m + 16] = make_float2(-2.0f * z, n);
        }
    }
    __syncthreads();

    const int wave   = t >> 5;             // row tile (cluster-2 rows r0..r0+15)
    const int lane   = t & 31;
    const int lane15 = lane & 15;
    const bool hi    = lane >= 16;
    const int r0     = wave * 16;

    const float2 af = apack[wave * 32 + lane];
    v2f afrag; afrag.x = af.x; afrag.y = af.y;

    // C layout: VGPR v, lanes 0-15 -> M=v (j=r0+v) ; lanes 16-31 -> M=v+8
    const unsigned int base2 =
        ((unsigned int)lane15 << 8) + (unsigned int)(r0 + (hi ? 8 : 0));

    float        bestVal  = __builtin_inff();
    unsigned int bestFlat = 0u;

#pragma unroll 4
    for (int ct = 0; ct < 16; ++ct) {
        const float4 bw = bpack[ct * 32 + lane];   // one ds_load_b128
        v2f bfrag; bfrag.x = bw.x; bfrag.y = bw.y;
        const float n1col = bw.z;                  // n1 of this lane's column i

        v8f acc = {};
        acc = __builtin_amdgcn_wmma_f32_16x16x4_f32(
            /*neg_a=*/false, afrag, /*neg_b=*/false, bfrag,
            /*c_mod=*/(short)0, acc, /*reuse_a=*/false, /*reuse_b=*/false);

        const unsigned int flatBase = base2 + ((unsigned int)ct << 12);
#pragma unroll
        for (int v = 0; v < 8; ++v) {
            const float d2 = acc[v] + n1col;
            const unsigned int fl = flatBase + (unsigned int)v;
            const bool lt = d2 < bestVal;          // strict: keep earliest flat idx
            bestFlat = lt ? fl : bestFlat;
            bestVal  = lt ? d2 : bestVal;
        }
    }

    // ---- pack (monotone value key, flat idx) once; u64-min = lexicographic argmin
    unsigned int bits = __float_as_uint(bestVal);
    unsigned int key  = bits ^ (((unsigned int)((int)bits >> 31)) | 0x80000000u);
    unsigned long long best =
        ((unsigned long long)key << 32) | (unsigned long long)bestFlat;

#pragma unroll
    for (int m = 16; m >= 1; m >>= 1) {
        unsigned long long o =
            (unsigned long long)__shfl_xor((long long)best, m, 32);
        best = (o < best) ? o : best;
    }
    if (lane == 0) wbest[wave] = best;
    __syncthreads();

    // ---- final reduce + 19-feature epilogue by thread 0
    if (t == 0) {
        unsigned long long g = wbest[0];
#pragma unroll
        for (int w = 1; w < 16; ++w) { if (wbest[w] < g) g = wbest[w]; }
        const unsigned int flat = (unsigned int)(g & 0xFFFFu);
        const int i = (int)(flat >> 8);    // cluster-1 voxel
        const int j = (int)(flat & 255u);  // cluster-2 voxel

        const float v1x = x1x[i], v1y = x1y[i], v1z = x1z[i];
        const float v2xv = x2x[j], v2yv = x2y[j], v2zv = x2z[j];
        float dx = v1x - v2xv, dy = v1y - v2yv, dz = v1z - v2zv;
        const float lend = sqrtf(dx * dx + dy * dy + dz * dz);
        const float denom = (lend > 0.0f) ? lend : 1.0f;
        dx /= denom; dy /= denom; dz /= denom;

        float* o = out + (size_t)e * 19;
        o[0] = v1x;  o[1] = v1y;  o[2] = v1z;
        o[3] = v2xv; o[4] = v2yv; o[5] = v2zv;
        o[6] = dx;   o[7] = dy;   o[8] = dz;
        o[9] = lend;
        o[10] = dx * dx; o[11] = dx * dy; o[12] = dx * dz;
        o[13] = dy * dx; o[14] = dy * dy; o[15] = dy * dz;
        o[16] = dz * dx; o[17] = dz * dy; o[18] = dz * dz;
    }
}

extern "C" void kernel_launch(void* const* d_in, const int* in_sizes, int n_in,
                              void* d_out, int out_size, void* d_ws, size_t ws_size,
                              hipStream_t stream) {
    const float* data       = (const float*)d_in[0];
    const int*   clusts     = (const int*)d_in[1];
    const int*   edge_index = (const int*)d_in[2];
    float*       out        = (float*)d_out;
    const int n_edge = in_sizes[2] / 2;   // edge_index is [2, E]

    clust_geo_edge_kernel<<<n_edge, TPB, 0, stream>>>(
        data, clusts, edge_index, out, n_edge);
}